// RandomErasing_59201829208647
// MI455X (gfx1250) — compile-verified
//
#include <hip/hip_runtime.h>
#include <hip/hip_bf16.h>

// Random erasing: out[b,h,w,c] = inside_rect(b,h,w) ? noise[...] : images[...]
// Shapes: B=64, H=224, W=224, C=3, float32.
//
// Strategy (MI455X, wave32):
//  - One block per (b, row): 14336 blocks x 224 threads (= 7 wave32s).
//    Rectangle params + row test are wave-uniform -> SALU; only the column
//    compare is per-lane.
//  - Per-lane *pointer* select (src = inside ? noise : images), then a single
//    96-bit load per pixel. This reads exactly one source stream per pixel
//    (~77 MB total traffic instead of ~116 MB), the binding constraint for a
//    0-FLOP select op on a 23.3 TB/s part.
//  - Non-temporal stores for the write-once output so it doesn't evict the
//    (replay-reused, L2-resident) input streams.

#define RE_B 64
#define RE_H 224
#define RE_W 224
#define RE_C 3

struct __attribute__((packed, aligned(4))) f3 { float x, y, z; };

__global__ __launch_bounds__(RE_W) void
RandomErasing_59201829208647_kernel(const float* __restrict__ images,
                                    const float* __restrict__ noise,
                                    const int* __restrict__ center_h,
                                    const int* __restrict__ center_w,
                                    const int* __restrict__ half_h,
                                    const int* __restrict__ half_w,
                                    float* __restrict__ out) {
    const int bh  = blockIdx.x;          // b*H + row
    const int b   = bh / RE_H;
    const int row = bh - b * RE_H;
    const int col = threadIdx.x;         // 0..223

    // Wave-uniform rectangle bounds (clipping is implicit in the compares,
    // since row/col are already within [0, H)/[0, W)).
    const int ch = center_h[b], cw = center_w[b];
    const int hh = half_h[b],   hw = half_w[b];
    const int r0 = ch - hh, r1 = ch + hh;
    const int c0 = cw - hw, c1 = cw + hw;

    const bool inside = (row >= r0) & (row < r1) & (col >= c0) & (col < c1);

    const size_t off = ((size_t)bh * RE_W + (size_t)col) * RE_C;

    // Per-lane base-pointer select: exactly one stream is read per pixel.
    const float* __restrict__ src = inside ? noise : images;

    const f3 v = *reinterpret_cast<const f3*>(src + off);

    // Write-once output: stream past the caches (NT temporal hint).
    __builtin_nontemporal_store(v.x, out + off + 0);
    __builtin_nontemporal_store(v.y, out + off + 1);
    __builtin_nontemporal_store(v.z, out + off + 2);
}

extern "C" void kernel_launch(void* const* d_in, const int* in_sizes, int n_in,
                              void* d_out, int out_size, void* d_ws, size_t ws_size,
                              hipStream_t stream) {
    (void)in_sizes; (void)n_in; (void)out_size; (void)d_ws; (void)ws_size;

    const float* images   = (const float*)d_in[0];
    const float* noise    = (const float*)d_in[1];
    const int*   center_h = (const int*)d_in[2];
    const int*   center_w = (const int*)d_in[3];
    const int*   half_h   = (const int*)d_in[4];
    const int*   half_w   = (const int*)d_in[5];
    float*       out      = (float*)d_out;

    const dim3 grid(RE_B * RE_H);   // 14336 blocks: one per (batch, row)
    const dim3 block(RE_W);         // 224 threads = 7 wave32s

    RandomErasing_59201829208647_kernel<<<grid, block, 0, stream>>>(
        images, noise, center_h, center_w, half_h, half_w, out);
}